// Amplituedro_90177133347655
// MI455X (gfx1250) — compile-verified
//
#include <hip/hip_runtime.h>
#include <hip/hip_bf16.h>

typedef float v2f __attribute__((ext_vector_type(2)));
typedef float v8f __attribute__((ext_vector_type(8)));

#define NUM_EXPERTS 64
#define D_MODEL     512
#define NTOK        65536
#define TOPK        8
#define TILE_M      16
#define TILES_PER_BLOCK 8    // 128 tokens / block
#define COLTILES_PER_WAVE 4  // 8 waves * 4 tiles * 16 cols = 512 cols

__global__ void eff_init_kernel(float* __restrict__ out) {
    if (threadIdx.x == 0 && blockIdx.x == 0)
        out[(size_t)NTOK * D_MODEL] = 0.0f;
}

__global__ __launch_bounds__(256)
void moe_path_wmma_kernel(const int*   __restrict__ eidx,   // [NTOK, TOPK] int32
                          const float* __restrict__ ew,     // [NTOK, TOPK] f32
                          const float* __restrict__ verts,  // [64, 512] f32
                          float*       __restrict__ out)    // [NTOK, 512] f32 + 1
{
    __shared__ float Wl[TILE_M * NUM_EXPERTS]; // dense per-token expert weights
    __shared__ float totl[TILE_M];
    __shared__ float scalel[TILE_M];
    __shared__ float norml[TILE_M];

    const int tid  = threadIdx.x;
    const int wave = tid >> 5;
    const int lane = tid & 31;
    const int n    = lane & 15;   // column within tile / M-row for A
    const int half = lane >> 4;   // 0 or 1 (selects K pair / row half)

    // ---- Preload B fragments (vertices slice) into registers, reused for all
    // token tiles handled by this block. Layout per ISA 7.12.2 for f32 B 4x16:
    // VGPR0: K = 0 | 2 (per half-wave), VGPR1: K = 1 | 3.
    v2f bfrag[COLTILES_PER_WAVE][16];
#pragma unroll
    for (int t = 0; t < COLTILES_PER_WAVE; ++t) {
        const int c0 = wave * 64 + t * 16;
#pragma unroll
        for (int s = 0; s < 16; ++s) {
            const int k0 = 4 * s + 2 * half;
            bfrag[t][s].x = verts[(size_t)k0 * D_MODEL + c0 + n];
            bfrag[t][s].y = verts[(size_t)(k0 + 1) * D_MODEL + c0 + n];
        }
    }

    float esum = 0.0f;
    const int tok0 = blockIdx.x * (TILES_PER_BLOCK * TILE_M);

    for (int tt = 0; tt < TILES_PER_BLOCK; ++tt) {
        __syncthreads();   // protect LDS reuse from previous iteration
        // zero the dense weight matrix + totals + norms
        for (int i = tid; i < TILE_M * NUM_EXPERTS; i += 256) Wl[i] = 0.0f;
        if (tid < TILE_M) { totl[tid] = 0.0f; norml[tid] = 0.0f; }
        __syncthreads();

        const int tbase = tok0 + tt * TILE_M;

        // scatter sparse top-k weights into dense 16x64 W (LDS float atomics)
        if (tid < TILE_M * TOPK) {
            const int m   = tid >> 3;
            const int k   = tid & 7;
            const int tok = tbase + m;
            const int e   = eidx[tok * TOPK + k];
            const float w = ew[tok * TOPK + k];
            if ((unsigned)e < NUM_EXPERTS) {
                atomicAdd(&Wl[m * NUM_EXPERTS + e], w);
                atomicAdd(&totl[m], w);
            }
        }
        __syncthreads();
        if (tid < TILE_M) {
            const float t = totl[tid];
            scalel[tid] = (t > 0.0f) ? (1.0f / t) : 1.0f;
        }
        __syncthreads();

        // ---- 16x16 tile GEMM: D = W(16x64) x verts(64x16), K in steps of 4
        v8f acc[COLTILES_PER_WAVE];
#pragma unroll
        for (int t = 0; t < COLTILES_PER_WAVE; ++t) acc[t] = (v8f)(0.0f);

#pragma unroll
        for (int s = 0; s < 16; ++s) {
            const int k0 = 4 * s + 2 * half;
            // A 16x4 f32 layout: lanes 0-15 & 16-31 both M=0..15;
            // VGPR0: K=0|2, VGPR1: K=1|3 -> one 64-bit LDS read.
            const v2f a = *(const v2f*)(&Wl[n * NUM_EXPERTS + k0]);
#pragma unroll
            for (int t = 0; t < COLTILES_PER_WAVE; ++t) {
                acc[t] = __builtin_amdgcn_wmma_f32_16x16x4_f32(
                    false, a, false, bfrag[t][s], (short)0, acc[t],
                    false, false);
            }
        }

        // ---- scale by 1/total_w, store, accumulate squared norms
        float np[8];
#pragma unroll
        for (int r = 0; r < 8; ++r) np[r] = 0.0f;
#pragma unroll
        for (int t = 0; t < COLTILES_PER_WAVE; ++t) {
            const int c0 = wave * 64 + t * 16;
#pragma unroll
            for (int r = 0; r < 8; ++r) {
                const int m = r + 8 * half;  // C/D layout: VGPR r -> rows r, r+8
                const float v = acc[t][r] * scalel[m];
                out[(size_t)(tbase + m) * D_MODEL + c0 + n] = v;
                np[r] += v * v;
            }
        }
#pragma unroll
        for (int r = 0; r < 8; ++r)
            atomicAdd(&norml[r + 8 * half], np[r]);
        __syncthreads();
        if (tid < TILE_M) esum += sqrtf(norml[tid]);
    }

    // efficiency = mean over all tokens of ||path||; accumulated globally
    if (tid < TILE_M)
        atomicAdd(out + (size_t)NTOK * D_MODEL, esum * (1.0f / (float)NTOK));
}

extern "C" void kernel_launch(void* const* d_in, const int* in_sizes, int n_in,
                              void* d_out, int out_size, void* d_ws, size_t ws_size,
                              hipStream_t stream) {
    (void)in_sizes; (void)n_in; (void)d_ws; (void)ws_size; (void)out_size;
    const int*   eidx  = (const int*)d_in[0];
    const float* ew    = (const float*)d_in[1];
    const float* verts = (const float*)d_in[2];
    float*       out   = (float*)d_out;

    eff_init_kernel<<<1, 32, 0, stream>>>(out);
    const int blocks = NTOK / (TILES_PER_BLOCK * TILE_M); // 512
    moe_path_wmma_kernel<<<blocks, 256, 0, stream>>>(eidx, ew, verts, out);
}